// GroupRCP_9603546874553
// MI455X (gfx1250) — compile-verified
//
#include <hip/hip_runtime.h>
#include <hip/hip_bf16.h>

typedef __attribute__((ext_vector_type(2))) float v2f;
typedef __attribute__((ext_vector_type(8))) float v8f;

#define HW 65536
#define CHN 128
#define NB 8

// ---- order-preserving float<->int map (for atomic min/max) ----
__device__ __forceinline__ int ford(float f) {
    int i = __float_as_int(f);
    return (i >= 0) ? i : (i ^ 0x7fffffff);
}
__device__ __forceinline__ float iford(int o) {
    int i = (o >= 0) ? o : (o ^ 0x7fffffff);
    return __int_as_float(i);
}

// ---------------- Kernel 1: per-(b,c) channel sums ----------------
__global__ void k_chan_sum(const float* __restrict__ x, float* __restrict__ sums) {
    __shared__ float red[256];
    const int bc = blockIdx.x;                 // 0..1023 = b*128 + c
    const float* p = x + (size_t)bc * HW;
    float s = 0.f;
    for (int i = threadIdx.x; i < HW; i += 256) s += p[i];
    red[threadIdx.x] = s;
    __syncthreads();
    for (int st = 128; st > 0; st >>= 1) {
        if (threadIdx.x < st) red[threadIdx.x] += red[threadIdx.x + st];
        __syncthreads();
    }
    if (threadIdx.x == 0) sums[bc] = red[0];   // sum ~ mean (same ordering)
}

// ---------------- Kernel 2: stable ranks + stats init ----------------
__global__ void k_rank(const float* __restrict__ sums, int* __restrict__ rank,
                       int* __restrict__ stats) {
    __shared__ float m[CHN];
    const int b = blockIdx.x, c = threadIdx.x;
    m[c] = sums[b * CHN + c];
    __syncthreads();
    const float mc = m[c];
    int r = 0;
    for (int j = 0; j < CHN; ++j) {
        float mj = m[j];
        r += (mj < mc) || (mj == mc && j < c);   // stable argsort position
    }
    rank[b * CHN + c] = r;
    if (b == 0 && c < 96)                        // 8 batches * 6 ch * {min,max}
        stats[c] = (c & 1) ? (int)0x80000000 : 0x7fffffff;
}

// ---------------- Kernel 3: WMMA band sums -> raw rcp (out ch 0..5) ----------------
// A-matrix rows: 0..2 prefix P32/P64/P96, 3 = total, 4..6 rank-set L32/L64/L96, 7..15 zero.
__global__ void __launch_bounds__(128) k_band_wmma(const float* __restrict__ x,
                                                   float* __restrict__ out,
                                                   const int* __restrict__ rank,
                                                   int* __restrict__ stats) {
    __shared__ float Alds[16 * CHN];            // 8 KB weight matrix (block-uniform)
    __shared__ int smin[6], smax[6];
    const int tid  = threadIdx.x;
    const int wave = tid >> 5, lane = tid & 31;
    const int tile = blockIdx.x * 4 + wave;     // one 16-pixel tile per wave
    const int p0   = tile << 4;
    const int b    = p0 >> 16;
    const int pixBase = p0 & (HW - 1);

    if (tid < 6) { smin[tid] = 0x7fffffff; smax[tid] = (int)0x80000000; }
    // build weight matrix once per block (branches only here, not in hot loop)
    for (int t = tid; t < 16 * CHN; t += 128) {
        const int m = t >> 7, c = t & 127;
        float w;
        if (m == 3)      w = 1.0f;
        else if (m < 3)  w = (c < 32 * (m + 1)) ? 1.0f : 0.0f;
        else if (m < 7)  w = (rank[b * CHN + c] < 32 * (m - 3)) ? 1.0f : 0.0f;
        else             w = 0.0f;
        Alds[t] = w;
    }
    __syncthreads();

    const int m    = lane & 15;                 // A row (and D row group)
    const int n    = lane & 15;                 // B column (pixel in tile)
    const int koff = (lane >> 4) << 1;          // K sub-offset 0 or 2
    const float* xb   = x + (size_t)b * CHN * HW + pixBase + n;
    const float* arow = &Alds[m * CHN + koff];

    v8f acc = {};
#pragma unroll
    for (int kk = 0; kk < CHN; kk += 4) {
        v2f A, Bv;
        A[0]  = arow[kk];
        A[1]  = arow[kk + 1];
        Bv[0] = xb[(size_t)(kk + koff) * HW];
        Bv[1] = xb[(size_t)(kk + koff + 1) * HW];
        acc = __builtin_amdgcn_wmma_f32_16x16x4_f32(
            false, A, false, Bv, (short)0, acc, false, false);
    }

    if (lane < 16) {
        // lanes 0..15 hold D rows 0..7 in acc[0..7] for pixel (pixBase+n)
        const float T = acc[3];
        float vals[6];
        // sorted_rcp (ch 0..2): high.mean - low.mean
        vals[0] = (T - acc[4]) / 96.f - acc[4] / 32.f;
        vals[1] = (T - acc[5]) / 64.f - acc[5] / 64.f;
        vals[2] = (T - acc[6]) / 32.f - acc[6] / 96.f;
        // serial_rcp (ch 3..5): light.mean - heavy.mean
        vals[3] = (T - acc[0]) / 96.f - acc[0] / 32.f;
        vals[4] = (T - acc[1]) / 64.f - acc[1] / 64.f;
        vals[5] = (T - acc[2]) / 32.f - acc[2] / 96.f;

        const size_t outBase = (size_t)b * 9 * HW + pixBase + n;
#pragma unroll
        for (int ch = 0; ch < 6; ++ch) {
            out[outBase + (size_t)ch * HW] = vals[ch];
            const int key = ford(vals[ch]);
            atomicMin(&smin[ch], key);
            atomicMax(&smax[ch], key);
        }
    }
    __syncthreads();
    if (tid < 6) {
        atomicMin(&stats[b * 12 + tid * 2 + 0], smin[tid]);
        atomicMax(&stats[b * 12 + tid * 2 + 1], smax[tid]);
    }
}

// ---------------- Kernel 4: per-pixel quartile partition (bitonic) ----------------
// Template-recursive network: guaranteed full unroll, v[] stays in VGPRs.
template <int K, int J>
__device__ __forceinline__ void ce_stage(float* v) {
#pragma unroll
    for (int i = 0; i < CHN; ++i) {
        const int ixj = i ^ J;
        if (ixj > i) {
            const bool up = ((i & K) == 0);
            const float a = v[i], c = v[ixj];
            const float lo = fminf(a, c), hi = fmaxf(a, c);
            v[i]   = up ? lo : hi;
            v[ixj] = up ? hi : lo;
        }
    }
}

template <int K, int J>
struct Merge {
    static __device__ __forceinline__ void run(float* v) {
        ce_stage<K, J>(v);
        Merge<K, (J >> 1)>::run(v);
    }
};
template <int K>
struct Merge<K, 0> {
    static __device__ __forceinline__ void run(float*) {}
};

__global__ void __launch_bounds__(128) k_sort_pw(const float* __restrict__ x,
                                                 float* __restrict__ out) {
    const int pixel = blockIdx.x * 128 + threadIdx.x;
    const int b = pixel >> 16, pix = pixel & (HW - 1);
    const float* p = x + (size_t)b * CHN * HW + pix;

    float v[CHN];
#pragma unroll
    for (int c = 0; c < CHN; ++c) v[c] = p[(size_t)c * HW];

    // Sort all 64-blocks (dir alternating) -> bitonic 128 sequence
    Merge<2, 1>::run(v);
    Merge<4, 2>::run(v);
    Merge<8, 4>::run(v);
    Merge<16, 8>::run(v);
    Merge<32, 16>::run(v);
    Merge<64, 32>::run(v);
    // First two passes of the final merge: partitions into exact rank-quartile
    // multisets at positions [0,32) [32,64) [64,96) [96,128).
    ce_stage<128, 64>(v);
    ce_stage<128, 32>(v);

    float s0 = 0.f, s1 = 0.f, s2 = 0.f, s3 = 0.f;
#pragma unroll
    for (int i = 0; i < 32; ++i) s0 += v[i];
#pragma unroll
    for (int i = 32; i < 64; ++i) s1 += v[i];
#pragma unroll
    for (int i = 64; i < 96; ++i) s2 += v[i];
#pragma unroll
    for (int i = 96; i < 128; ++i) s3 += v[i];

    const float Q32 = s0, Q64 = s0 + s1, Q96 = s0 + s1 + s2;
    const float T = Q96 + s3;
    const size_t ob = (size_t)b * 9 * HW + pix;
    out[ob + (size_t)6 * HW] = Q32 / 32.f - (T - Q32) / 96.f;
    out[ob + (size_t)7 * HW] = Q64 / 64.f - (T - Q64) / 64.f;
    out[ob + (size_t)8 * HW] = Q96 / 96.f - (T - Q96) / 32.f;
}

// ---------------- Kernel 5: in-place min-max normalize (out ch 0..5) ----------------
__global__ void k_norm(float* __restrict__ out, const int* __restrict__ stats) {
    const int idx = blockIdx.x * 256 + threadIdx.x;   // < 8*6*65536
    const int b = idx / (6 * HW);
    const int r = idx - b * (6 * HW);
    const int ch = r >> 16;
    const int pix = r & (HW - 1);
    const float mn = iford(stats[b * 12 + ch * 2 + 0]);
    const float mx = iford(stats[b * 12 + ch * 2 + 1]);
    const size_t o = ((size_t)b * 9 + ch) * HW + pix;
    out[o] = (out[o] - mn) / (mx - mn + 1e-8f);
}

extern "C" void kernel_launch(void* const* d_in, const int* in_sizes, int n_in,
                              void* d_out, int out_size, void* d_ws, size_t ws_size,
                              hipStream_t stream) {
    const float* x = (const float*)d_in[0];
    float* out = (float*)d_out;

    float* sums  = (float*)d_ws;                // 1024 floats
    int*   rank  = (int*)d_ws + 1024;           // 1024 ints
    int*   stats = (int*)d_ws + 2048;           // 96 ints

    k_chan_sum<<<NB * CHN, 256, 0, stream>>>(x, sums);
    k_rank<<<NB, CHN, 0, stream>>>(sums, rank, stats);
    k_band_wmma<<<(NB * HW / 16) / 4, 128, 0, stream>>>(x, out, rank, stats);
    k_sort_pw<<<NB * HW / 128, 128, 0, stream>>>(x, out);
    k_norm<<<NB * 6 * HW / 256, 256, 0, stream>>>(out, stats);
}